// LennardJones_40544491274907
// MI455X (gfx1250) — compile-verified
//
#include <hip/hip_runtime.h>
#include <hip/hip_bf16.h>

// Lennard-Jones per-edge energy -> per-atom scatter-add (half-list).
//   r2 = |d|^2 ; sr6 = r2^-3 (SIGMA=1) ; e = 4*eps*(sr6^2 - sr6) - SHIFT
//   out[first] += e/2 ; out[second] += e/2
//
// MI455X roofline: ~180 MB read-once streams (~8 us @ 23.3 TB/s) + 12.8M
// non-returning fp32 atomics into a 400 KB accumulator resident in the
// 192 MB L2. No matrix structure -> WMMA inapplicable. Optimization targets:
//   * all streams via 128-bit non-temporal loads (4 edges/thread)
//   * global_atomic_add_f32 without return (STOREcnt path, L2 RMW units)
//   * wave32-friendly 256-thread blocks, enough waves to hide scatter latency

typedef float v4f __attribute__((ext_vector_type(4)));
typedef int   v4i __attribute__((ext_vector_type(4)));

#ifndef LJ_EPSILON
#define LJ_EPSILON 1.0f
#endif

static constexpr float kSigma  = 1.0f;
static constexpr float kCutoff = 5.0f;
static constexpr float kInvC2  = (kSigma / kCutoff) * (kSigma / kCutoff);
static constexpr float kInvC6  = kInvC2 * kInvC2 * kInvC2;
static constexpr float kShift  = 4.0f * LJ_EPSILON * (kInvC6 * kInvC6 - kInvC6);

__device__ __forceinline__ float lj_half_energy(float x, float y, float z) {
    float r2   = x * x + y * y + z * z;
    float inv2 = 1.0f / r2;
    float sr6  = inv2 * inv2 * inv2;
    float e    = 4.0f * LJ_EPSILON * (sr6 * sr6 - sr6) - kShift;
    return 0.5f * e;
}

__global__ __launch_bounds__(256) void lj_zero_kernel(float* __restrict__ out, int n) {
    int i = blockIdx.x * blockDim.x + threadIdx.x;
    if (i < n) out[i] = 0.0f;
}

__global__ __launch_bounds__(256) void lj_edge_kernel(
    const float* __restrict__ dist,        // [E,3] row-major
    const int*   __restrict__ first_atom,  // [E]
    const int*   __restrict__ second_atom, // [E]
    float*       __restrict__ energy,      // [n_atoms]
    int n_edges)
{
    const int t    = blockIdx.x * blockDim.x + threadIdx.x;
    const int base = t * 4;
    if (base >= n_edges) return;

    if (base + 3 < n_edges) {
        // Fast path: 4 edges = 48 B of distances = 3 aligned b128 NT loads,
        // plus one b128 NT load from each index array.
        const v4f* dp = (const v4f*)(dist + 12ull * (unsigned)t);
        v4f d0 = __builtin_nontemporal_load(dp + 0);   // e0.xyz e1.x
        v4f d1 = __builtin_nontemporal_load(dp + 1);   // e1.yz  e2.xy
        v4f d2 = __builtin_nontemporal_load(dp + 2);   // e2.z   e3.xyz
        v4i ia = __builtin_nontemporal_load((const v4i*)(first_atom  + base));
        v4i ib = __builtin_nontemporal_load((const v4i*)(second_atom + base));

        float h0 = lj_half_energy(d0.x, d0.y, d0.z);
        float h1 = lj_half_energy(d0.w, d1.x, d1.y);
        float h2 = lj_half_energy(d1.z, d1.w, d2.x);
        float h3 = lj_half_energy(d2.y, d2.z, d2.w);

        unsafeAtomicAdd(energy + ia.x, h0);
        unsafeAtomicAdd(energy + ib.x, h0);
        unsafeAtomicAdd(energy + ia.y, h1);
        unsafeAtomicAdd(energy + ib.y, h1);
        unsafeAtomicAdd(energy + ia.z, h2);
        unsafeAtomicAdd(energy + ib.z, h2);
        unsafeAtomicAdd(energy + ia.w, h3);
        unsafeAtomicAdd(energy + ib.w, h3);
    } else {
        // Tail: scalar bounds-checked path (empty when E % 4 == 0).
        for (int i = base; i < n_edges; ++i) {
            const float* dp = dist + 3ull * (unsigned)i;
            float x = __builtin_nontemporal_load(dp + 0);
            float y = __builtin_nontemporal_load(dp + 1);
            float z = __builtin_nontemporal_load(dp + 2);
            int a = __builtin_nontemporal_load(first_atom + i);
            int b = __builtin_nontemporal_load(second_atom + i);
            float h = lj_half_energy(x, y, z);
            unsafeAtomicAdd(energy + a, h);
            unsafeAtomicAdd(energy + b, h);
        }
    }
}

extern "C" void kernel_launch(void* const* d_in, const int* in_sizes, int n_in,
                              void* d_out, int out_size, void* d_ws, size_t ws_size,
                              hipStream_t stream) {
    const float* dist        = (const float*)d_in[0];   // [E,3] float32
    const int*   first_atom  = (const int*)d_in[1];     // [E] int
    const int*   second_atom = (const int*)d_in[2];     // [E] int
    float*       energy      = (float*)d_out;           // [n_atoms] float32

    const int n_edges = in_sizes[0] / 3;
    const int n_atoms = out_size;

    // d_out is poisoned by the harness: zero it first (same stream => ordered).
    {
        int blocks = (n_atoms + 255) / 256;
        lj_zero_kernel<<<blocks, 256, 0, stream>>>(energy, n_atoms);
    }
    {
        int edges_per_block = 256 * 4;
        int blocks = (n_edges + edges_per_block - 1) / edges_per_block;
        lj_edge_kernel<<<blocks, 256, 0, stream>>>(dist, first_atom, second_atom,
                                                   energy, n_edges);
    }
}